// TopkGCN_20744692039844
// MI455X (gfx1250) — compile-verified
//
#include <hip/hip_runtime.h>
#include <math.h>
#include <stdint.h>

#define FHH 64
#define NG 256
#define NEG_INF -3.402823466e38f
#define BPAD 4          // LDS row padding (floats) to spread banks
#define BROW (64 + BPAD)

typedef __attribute__((ext_vector_type(2))) float v2f;
typedef __attribute__((ext_vector_type(8))) float v8f;

static inline int cdiv(int a, int b) { return (a + b - 1) / b; }

// ---------------- WMMA GEMM: C[n x P] = A[n x K] @ B[K x P] (+bias, optional relu)
// Requirements: n % 128 == 0, K % 64 == 0, P % 64 == 0.
// Block = 256 threads (8 waves) -> 128 rows x 64 cols of C.
// Each wave: 16 rows x four 16x16 WMMA tiles. B staged to LDS with
// gfx1250 async-to-LDS loads (ASYNCcnt), A fragments from global.
__global__ void wmma_mm_kernel(const float* __restrict__ A, const float* __restrict__ B,
                               const float* __restrict__ bias, float* __restrict__ C,
                               int n, int K, int P, int do_relu)
{
    __shared__ float Bsh[64 * BROW];

    int colGroups = P >> 6;                      // groups of 64 columns
    int rowBlock  = blockIdx.x / colGroups;
    int colBase   = (blockIdx.x - rowBlock * colGroups) << 6;

    int waveId = threadIdx.x >> 5;
    int lane   = threadIdx.x & 31;
    int l15    = lane & 15;
    int hi     = lane >> 4;                      // 0: K pair {0,1}; 1: {2,3} within group of 4
    int kgrp   = hi << 1;

    int rowTop = rowBlock * 128 + waveId * 16;   // 16 rows owned by this wave
    const float* arow = A + (size_t)(rowTop + l15) * K;
    __builtin_prefetch(arow, 0, 0);              // global_prefetch_b8 warm-up

    v8f acc0 = {}, acc1 = {}, acc2 = {}, acc3 = {};

    for (int kk = 0; kk < K; kk += 64) {
        if (kk) __syncthreads();                 // previous chunk fully consumed
        // ---- async stage B[kk:kk+64, colBase:colBase+64] -> LDS (b128 per lane)
        #pragma unroll
        for (int i = 0; i < 4; ++i) {
            int c    = threadIdx.x + (i << 8);   // 0..1023 16-byte chunks
            int brow = c >> 4;                   // 0..63
            int quad = c & 15;                   // 16B chunk within row
            const float* gsrc = B + (size_t)(kk + brow) * P + colBase + (quad << 2);
            unsigned ldst = (unsigned)(uintptr_t)&Bsh[brow * BROW + (quad << 2)];
            asm volatile("global_load_async_to_lds_b128 %0, %1, off"
                         :: "v"(ldst), "v"(gsrc)
                         : "memory");
        }
        asm volatile("s_wait_asynccnt 0x0" ::: "memory");
        __syncthreads();

        // ---- 16 k-steps x 4 WMMA tiles, A fragment reused across tiles
        #pragma unroll 4
        for (int k0 = kgrp; k0 < 64; k0 += 4) {
            v2f av;
            av.x = arow[kk + k0];
            av.y = arow[kk + k0 + 1];
            const float* b0 = &Bsh[k0 * BROW];
            const float* b1 = &Bsh[(k0 + 1) * BROW];
            v2f bv0, bv1, bv2, bv3;
            bv0.x = b0[l15];        bv0.y = b1[l15];
            bv1.x = b0[16 + l15];   bv1.y = b1[16 + l15];
            bv2.x = b0[32 + l15];   bv2.y = b1[32 + l15];
            bv3.x = b0[48 + l15];   bv3.y = b1[48 + l15];
            acc0 = __builtin_amdgcn_wmma_f32_16x16x4_f32(false, av, false, bv0, (short)0, acc0, false, false);
            acc1 = __builtin_amdgcn_wmma_f32_16x16x4_f32(false, av, false, bv1, (short)0, acc1, false, false);
            acc2 = __builtin_amdgcn_wmma_f32_16x16x4_f32(false, av, false, bv2, (short)0, acc2, false, false);
            acc3 = __builtin_amdgcn_wmma_f32_16x16x4_f32(false, av, false, bv3, (short)0, acc3, false, false);
        }
    }

    // ---- writeback: lanes 0-15 -> M=r, lanes 16-31 -> M=r+8 within the tile
    int rbase = rowTop + (hi << 3);
    #pragma unroll
    for (int t = 0; t < 4; ++t) {
        v8f acc = (t == 0) ? acc0 : (t == 1) ? acc1 : (t == 2) ? acc2 : acc3;
        int col = colBase + (t << 4) + l15;
        float badd = bias ? bias[col] : 0.0f;
        #pragma unroll
        for (int r = 0; r < 8; ++r) {
            float v = acc[r] + badd;
            if (do_relu) v = fmaxf(v, 0.0f);
            C[(size_t)(rbase + r) * P + col] = v;
        }
    }
}

// ---------------- fills
__global__ void fill_f32(float* p, float v, int n) {
    int i = blockIdx.x * blockDim.x + threadIdx.x;
    if (i < n) p[i] = v;
}
__global__ void fill_i32(int* p, int v, int n) {
    int i = blockIdx.x * blockDim.x + threadIdx.x;
    if (i < n) p[i] = v;
}

// ---------------- degree accumulation over edges (deg pre-filled with 1.0 self-loop)
__global__ void deg_edges_kernel(const int* __restrict__ col, const float* __restrict__ ew,
                                 float* __restrict__ deg, int E) {
    int e = blockIdx.x * blockDim.x + threadIdx.x;
    if (e >= E) return;
    float w = ew[e];
    if (w != 0.0f) atomicAdd(&deg[col[e]], w);
}
__global__ void dinv_kernel(float* deg, int n) {
    int i = blockIdx.x * blockDim.x + threadIdx.x;
    if (i < n) deg[i] = rsqrtf(deg[i]);      // in place: deg -> dinv
}

// ---------------- out = dinv^2 * xw + bias  (self-loop term, full overwrite)
__global__ void selfloop_bias_kernel(const float* __restrict__ xw, const float* __restrict__ dinv,
                                     const float* __restrict__ b, float* __restrict__ out, int n) {
    int idx = blockIdx.x * blockDim.x + threadIdx.x;
    if (idx >= n * FHH) return;
    int i = idx >> 6, f = idx & 63;
    float di = dinv[i];
    out[idx] = di * di * xw[idx] + b[f];
}

// ---------------- agg[col] += dinv[row]*ew*dinv[col] * xw[row], one thread per (edge,f)
__global__ void scatter_edges_kernel(const int* __restrict__ row, const int* __restrict__ col,
                                     const float* __restrict__ ew, const float* __restrict__ dinv,
                                     const float* __restrict__ xw, float* __restrict__ out, int E) {
    int idx = blockIdx.x * blockDim.x + threadIdx.x;   // E*64 <= 2^26, fits int
    if (idx >= E * FHH) return;
    int e = idx >> 6, f = idx & 63;
    float w = ew[e];
    if (w == 0.0f) return;
    int r = row[e], c = col[e];
    float coef = dinv[r] * w * dinv[c];
    atomicAdd(&out[(size_t)c * FHH + f], coef * xw[(size_t)r * FHH + f]);
}

// ---------------- BatchNorm stats: sums[0:64]=sum, sums[64:128]=sumsq (pre-zeroed)
__global__ void bn_stats_kernel(const float* __restrict__ x, float* __restrict__ sums, int n) {
    __shared__ float ls[FHH], lq[FHH];
    int t = threadIdx.x;
    if (t < FHH) { ls[t] = 0.0f; lq[t] = 0.0f; }
    __syncthreads();
    int total = n * FHH;
    int f = t & 63;                          // stride (blockDim*gridDim) is a multiple of 64
    float s = 0.0f, q = 0.0f;
    for (int i = blockIdx.x * blockDim.x + t; i < total; i += gridDim.x * blockDim.x) {
        float v = x[i];
        s += v; q += v * v;
    }
    atomicAdd(&ls[f], s);
    atomicAdd(&lq[f], q);
    __syncthreads();
    if (t < FHH) { atomicAdd(&sums[t], ls[t]); atomicAdd(&sums[FHH + t], lq[t]); }
}

// ---------------- BN apply + ReLU (in place)
__global__ void bn_apply_kernel(float* __restrict__ x, const float* __restrict__ sums,
                                const float* __restrict__ gam, const float* __restrict__ bet, int n) {
    int idx = blockIdx.x * blockDim.x + threadIdx.x;
    if (idx >= n * FHH) return;
    int f = idx & 63;
    float inv_n = 1.0f / (float)n;
    float mu  = sums[f] * inv_n;
    float var = sums[FHH + f] * inv_n - mu * mu;
    float v = (x[idx] - mu) * rsqrtf(var + 1e-5f) * gam[f] + bet[f];
    x[idx] = v > 0.0f ? v : 0.0f;
}

// ---------------- attention score: tanh((h . a)/||a||)
__global__ void score_kernel(const float* __restrict__ h, const float* __restrict__ a,
                             float* __restrict__ score, int n) {
    int i = blockIdx.x * blockDim.x + threadIdx.x;
    if (i >= n) return;
    float d = 0.0f, na = 0.0f;
    const float* hr = h + (size_t)i * FHH;
    #pragma unroll
    for (int f = 0; f < FHH; ++f) { d += hr[f] * a[f]; na += a[f] * a[f]; }
    score[i] = tanhf(d * rsqrtf(na));
}

// ---------------- per-graph top-k via bitonic sort over 256 padded slots
__global__ void topk_kernel(const float* __restrict__ score, float* __restrict__ vals,
                            int* __restrict__ perm, int* __restrict__ newid,
                            int n_in, int k) {
    __shared__ float s[256];
    __shared__ int   id[256];
    int g = blockIdx.x, t = threadIdx.x;
    s[t]  = (t < n_in) ? score[g * n_in + t] : NEG_INF;
    id[t] = t;
    __syncthreads();
    for (int size = 2; size <= 256; size <<= 1) {
        for (int stride = size >> 1; stride > 0; stride >>= 1) {
            int p = t ^ stride;
            if (p > t) {
                bool desc = ((t & size) == 0);
                bool sw = desc ? (s[t] < s[p]) : (s[t] > s[p]);
                if (sw) {
                    float tv = s[t]; s[t] = s[p]; s[p] = tv;
                    int ti = id[t]; id[t] = id[p]; id[p] = ti;
                }
            }
            __syncthreads();
        }
    }
    if (t < k) {
        int node = g * n_in + id[t];
        int j = g * k + t;
        perm[j] = node;
        vals[j] = s[t];
        newid[node] = j;
    }
}

// ---------------- gather + gate kept nodes: hk[j] = h[perm[j]] * vals[j]
__global__ void gather_kernel(const float* __restrict__ h, const int* __restrict__ perm,
                              const float* __restrict__ vals, float* __restrict__ hk, int nk) {
    int idx = blockIdx.x * blockDim.x + threadIdx.x;
    if (idx >= nk * FHH) return;
    int j = idx >> 6, f = idx & 63;
    hk[idx] = h[(size_t)perm[j] * FHH + f] * vals[j];
}

// ---------------- edge remap with keep mask (dropped edges -> node 0, weight 0)
__global__ void remap_edges_kernel(const int* __restrict__ row, const int* __restrict__ col,
                                   const float* __restrict__ ew, const int* __restrict__ newid,
                                   int* __restrict__ row2, int* __restrict__ col2,
                                   float* __restrict__ ew2, int E) {
    int e = blockIdx.x * blockDim.x + threadIdx.x;
    if (e >= E) return;
    int nr = newid[row[e]], nc = newid[col[e]];
    float w = ew[e];
    bool keep = (nr >= 0) && (nc >= 0) && (w > 0.0f);
    row2[e] = keep ? nr : 0;
    col2[e] = keep ? nc : 0;
    ew2[e]  = keep ? w : 0.0f;
}

// ---------------- per-graph mean||max pool, accumulated into flatten
__global__ void pool_kernel(const float* __restrict__ hk, float* __restrict__ flatten, int k) {
    int g = blockIdx.x, f = threadIdx.x;     // 64 threads
    float sm = 0.0f, mx = NEG_INF;
    const float* base = hk + (size_t)g * k * FHH + f;
    for (int i = 0; i < k; ++i) {
        float v = base[(size_t)i * FHH];
        sm += v;
        mx = fmaxf(mx, v);
    }
    flatten[g * 128 + f]      += sm / (float)k;
    flatten[g * 128 + 64 + f] += mx;
}

// ---------------- final 256 -> 10 layer
__global__ void mlp2_kernel(const float* __restrict__ hdn, const float* __restrict__ D2w,
                            const float* __restrict__ D2b, float* __restrict__ out) {
    int idx = blockIdx.x * blockDim.x + threadIdx.x;
    if (idx >= NG * 10) return;
    int g = idx / 10, c = idx - g * 10;
    float s = D2b[c];
    const float* hr = hdn + (size_t)g * 256;
    for (int j = 0; j < 256; ++j) s += hr[j] * D2w[j * 10 + c];
    out[idx] = s;
}

extern "C" void kernel_launch(void* const* d_in, const int* in_sizes, int n_in_cnt,
                              void* d_out, int out_size, void* d_ws, size_t ws_size,
                              hipStream_t stream) {
    (void)n_in_cnt; (void)out_size; (void)ws_size;
    const float* inputs     = (const float*)d_in[0];
    const int*   edge_index = (const int*)  d_in[1];
    const float* edge_w_in  = (const float*)d_in[3];
    const float* Wb[3]  = { (const float*)d_in[4],  (const float*)d_in[9],  (const float*)d_in[14] };
    const float* bb[3]  = { (const float*)d_in[5],  (const float*)d_in[10], (const float*)d_in[15] };
    const float* gb[3]  = { (const float*)d_in[6],  (const float*)d_in[11], (const float*)d_in[16] };
    const float* beb[3] = { (const float*)d_in[7],  (const float*)d_in[12], (const float*)d_in[17] };
    const float* ab[3]  = { (const float*)d_in[8],  (const float*)d_in[13], (const float*)d_in[18] };
    const float* D1w = (const float*)d_in[19];
    const float* D1b = (const float*)d_in[20];
    const float* D2w = (const float*)d_in[21];
    const float* D2b = (const float*)d_in[22];
    float* out = (float*)d_out;

    const int N = in_sizes[0] / FHH;          // 65536
    const int E = in_sizes[3];                // 1048576
    const int* row0 = edge_index;
    const int* col0 = edge_index + E;

    // ---- carve workspace
    char* curp = (char*)d_ws;
    auto carve = [&](size_t bytes) -> void* {
        void* r = (void*)curp;
        curp += (bytes + 255) & ~(size_t)255;
        return r;
    };
    size_t bigBytes = (size_t)N * FHH * sizeof(float);
    float* big0  = (float*)carve(bigBytes);
    float* big1  = (float*)carve(bigBytes);   // fixed role: conv/BN output
    float* big2  = (float*)carve(bigBytes);
    float* deg   = (float*)carve((size_t)N * sizeof(float));
    float* score = (float*)carve((size_t)N * sizeof(float));
    float* vals  = (float*)carve((size_t)N * sizeof(float));
    int*   perm  = (int*)  carve((size_t)N * sizeof(int));
    int*   newid = (int*)  carve((size_t)N * sizeof(int));
    int*   rowA  = (int*)  carve((size_t)E * sizeof(int));
    int*   colA  = (int*)  carve((size_t)E * sizeof(int));
    float* ewA   = (float*)carve((size_t)E * sizeof(float));
    int*   rowB  = (int*)  carve((size_t)E * sizeof(int));
    int*   colB  = (int*)  carve((size_t)E * sizeof(int));
    float* ewB   = (float*)carve((size_t)E * sizeof(float));
    float* sums  = (float*)carve(128 * sizeof(float));
    float* flatten = (float*)carve((size_t)NG * 128 * sizeof(float));
    float* hdn     = (float*)carve((size_t)NG * 256 * sizeof(float));

    fill_f32<<<cdiv(NG * 128, 256), 256, 0, stream>>>(flatten, 0.0f, NG * 128);

    int n_in = N / NG;                        // 256 nodes per graph initially
    const float* h_cur = inputs;
    const int* rc = row0; const int* cc = col0; const float* ec = edge_w_in;
    int* rn[3] = { rowA, rowB, rowA };
    int* cn[3] = { colA, colB, colA };
    float* en[3] = { ewA, ewB, ewA };

    for (int b = 0; b < 3; ++b) {
        int n = NG * n_in;                    // 65536 / 52480 / 41984 (all % 128 == 0)
        int k = (4 * n_in + 4) / 5;           // ceil(0.8 * n_in): 205 / 164 / 132
        int nk = NG * k;

        // deterministic buffer roles, no aliasing:
        //   out = big1 always; xw and hk share big0 (even b) or big2 (odd b);
        //   hk may overwrite xw because xw is dead after the edge scatter.
        float* Bxw  = (b & 1) ? big2 : big0;
        float* Bout = big1;
        float* Bhk  = Bxw;

        // xw = h_cur @ W  (fp32 WMMA, B staged via async-to-LDS)
        wmma_mm_kernel<<<(n / 128) * (FHH / 64), 256, 0, stream>>>(
            h_cur, Wb[b], nullptr, Bxw, n, FHH, FHH, 0);
        // degree (self-loop 1.0) -> dinv
        fill_f32<<<cdiv(n, 256), 256, 0, stream>>>(deg, 1.0f, n);
        deg_edges_kernel<<<cdiv(E, 256), 256, 0, stream>>>(cc, ec, deg, E);
        dinv_kernel<<<cdiv(n, 256), 256, 0, stream>>>(deg, n);
        // out = dinv^2*xw + bias, then edge scatter (one thread per edge-feature)
        selfloop_bias_kernel<<<cdiv(n * FHH, 256), 256, 0, stream>>>(Bxw, deg, bb[b], Bout, n);
        scatter_edges_kernel<<<cdiv(E * FHH, 256), 256, 0, stream>>>(rc, cc, ec, deg, Bxw, Bout, E);
        // BatchNorm (training stats) + ReLU
        fill_f32<<<1, 128, 0, stream>>>(sums, 0.0f, 128);
        bn_stats_kernel<<<512, 256, 0, stream>>>(Bout, sums, n);
        bn_apply_kernel<<<cdiv(n * FHH, 256), 256, 0, stream>>>(Bout, sums, gb[b], beb[b], n);
        // top-k pooling
        score_kernel<<<cdiv(n, 256), 256, 0, stream>>>(Bout, ab[b], score, n);
        fill_i32<<<cdiv(n, 256), 256, 0, stream>>>(newid, -1, n);
        topk_kernel<<<NG, 256, 0, stream>>>(score, vals, perm, newid, n_in, k);
        gather_kernel<<<cdiv(nk * FHH, 256), 256, 0, stream>>>(Bout, perm, vals, Bhk, nk);
        remap_edges_kernel<<<cdiv(E, 256), 256, 0, stream>>>(rc, cc, ec, newid,
                                                             rn[b], cn[b], en[b], E);
        pool_kernel<<<NG, 64, 0, stream>>>(Bhk, flatten, k);

        h_cur = Bhk;
        rc = rn[b]; cc = cn[b]; ec = en[b];
        n_in = k;
    }

    // MLP layer 1: relu(flatten[256x128] @ D1w[128x256] + D1b) via fp32 WMMA
    wmma_mm_kernel<<<(NG / 128) * (256 / 64), 256, 0, stream>>>(
        flatten, D1w, D1b, hdn, NG, 128, 256, 1);
    // MLP layer 2: hdn @ D2w + D2b
    mlp2_kernel<<<cdiv(NG * 10, 256), 256, 0, stream>>>(hdn, D2w, D2b, out);
}